// UniModel_29961691857261
// MI455X (gfx1250) — compile-verified
//
#include <hip/hip_runtime.h>
#include <math.h>

// Problem sizes (compile-time constants from the reference).
#define T_DIM 4096
#define H_DIM 2048
#define O_DIM 512

typedef __attribute__((ext_vector_type(2))) float    v2f;
typedef __attribute__((ext_vector_type(8))) float    v8f;
typedef __attribute__((ext_vector_type(4))) unsigned v4u;
typedef __attribute__((ext_vector_type(8))) int      v8i;
typedef __attribute__((ext_vector_type(4))) int      v4i;

static __device__ __forceinline__ v8f wmma4(v2f a, v2f b, v8f c) {
    return __builtin_amdgcn_wmma_f32_16x16x4_f32(
        /*neg_a=*/false, a, /*neg_b=*/false, b,
        /*c_mod=*/(short)0, c, /*reuse_a=*/false, /*reuse_b=*/false);
}

// ---------------------------------------------------------------------------
// FP32 WMMA GEMM:  C[M,N] = A[M,K] * B[N,K]^T + bias[N]
// V_WMMA_F32_16X16X4_F32, one wave per 32x32 C tile (2x2 register blocking:
// each A/B fragment feeds two WMMAs -> 2x the FLOP/byte of a 16x16 tile).
//
// A-fragment layout (ISA 7.12.2, 32-bit A 16x4): lanes 0-15 hold M=lane,
// v0=K0,v1=K1; lanes 16-31 hold M=lane-16, v0=K2,v1=K3.
// B-fragment mirrors A with M<->N. C/D: VGPR r -> row r+8*(lane>=16),
// col = lane&15.
// ---------------------------------------------------------------------------
__global__ void gemm16x16x4(const float* __restrict__ A,   // M x K row-major
                            const float* __restrict__ B,   // N x K row-major
                            const float* __restrict__ bias,// N
                            float* __restrict__ C,         // M x N row-major
                            int M, int N, int K)
{
    const int lane = threadIdx.x & 31;
    const int wave = threadIdx.x >> 5;        // 4 waves/block, adjacent N tiles
    const int bm   = blockIdx.x;              // 32-row tile index
    const int bn   = blockIdx.y * 4 + wave;   // 32-col tile index
    const int nl   = lane & 15;
    const int hi   = lane >> 4;
    const int k0   = hi * 2;

    const float* a0 = A + (size_t)(bm * 32 + nl)      * K + k0;
    const float* a1 = A + (size_t)(bm * 32 + 16 + nl) * K + k0;
    const float* b0 = B + (size_t)(bn * 32 + nl)      * K + k0;
    const float* b1 = B + (size_t)(bn * 32 + 16 + nl) * K + k0;

    v8f acc00 = {}, acc01 = {}, acc10 = {}, acc11 = {};
#pragma unroll 4
    for (int k = 0; k < K; k += 4) {
        v2f va0 = *(const v2f*)(a0 + k);   // 8B-aligned (k even)
        v2f va1 = *(const v2f*)(a1 + k);
        v2f vb0 = *(const v2f*)(b0 + k);
        v2f vb1 = *(const v2f*)(b1 + k);
        acc00 = wmma4(va0, vb0, acc00);
        acc01 = wmma4(va0, vb1, acc01);
        acc10 = wmma4(va1, vb0, acc10);
        acc11 = wmma4(va1, vb1, acc11);
    }

    const int   col0 = bn * 32 + nl, col1 = col0 + 16;
    const float bv0  = bias[col0],   bv1  = bias[col1];
#pragma unroll
    for (int r = 0; r < 8; ++r) {
        int row0 = bm * 32 + r + 8 * hi;
        int row1 = row0 + 16;
        C[(size_t)row0 * N + col0] = acc00[r] + bv0;
        C[(size_t)row0 * N + col1] = acc01[r] + bv1;
        C[(size_t)row1 * N + col0] = acc10[r] + bv0;
        C[(size_t)row1 * N + col1] = acc11[r] + bv1;
    }
}

// ---------------------------------------------------------------------------
// Persistent RNN recurrence: h_t = tanh(xp[t] + W_hh h_{t-1} + b_hh)
// 64 WGs x 256 threads; each WG pins 32 rows of W_hh in LDS at pitch 2056
// floats (row r, sub s -> bank (8r+s)&63: conflict-free). Weight preload via
// the Tensor Data Mover (iterated 1x2048 tile, 32 iterations, LDS increment
// = pitch). Grid barrier per step: release-add / acquire-spin on a counter.
// outs[t] overwrites xp[t] in place.
// ---------------------------------------------------------------------------
#define NWG          64
#define RNN_THREADS  256
#define ROWS_PER_WG  (H_DIM / NWG)      // 32
#define W_PITCH      2056               // 2048 + 8 floats, bank-conflict-free
#define RNN_LDS_BYTES ((ROWS_PER_WG * W_PITCH + H_DIM + RNN_THREADS) * 4)

__global__ void rnn_persistent(float* __restrict__ xpo,        // T x H: in xp, out h_t
                               const float* __restrict__ Whh,  // H x H
                               const float* __restrict__ bhh,  // H
                               const float* __restrict__ h0,   // H
                               float* __restrict__ hbuf,       // 2 x H double buffer
                               unsigned* __restrict__ counter)
{
    extern __shared__ float smem[];
    float* wsl = smem;                             // ROWS_PER_WG * W_PITCH
    float* hs  = smem + ROWS_PER_WG * W_PITCH;     // H
    float* red = hs + H_DIM;                       // RNN_THREADS

    const int tid = threadIdx.x;
    const int wg  = blockIdx.x;

    // --- TDM preload of this WG's W_hh slice (wave 0 issues one descriptor) ---
    // amdgpu-toolchain (clang-23) 6-arg builtin form:
    //   (v4u g0, v8i g1, v4i g2, v4i g3, v8i aux, i32 cpol)
    if (tid < 32) {
        // LDS flat address: low 32 bits == LDS byte offset (aperture layout).
        unsigned lds_base = (unsigned)(uintptr_t)wsl;
        unsigned long long ga =
            (unsigned long long)(uintptr_t)(Whh + (size_t)wg * ROWS_PER_WG * H_DIM);
        // D# Group 0: count=1 | lds_addr | global_addr[56:0] | type=2.
        v4u g0 = { 1u,
                   lds_base,
                   (unsigned)(ga & 0xffffffffu),
                   (unsigned)((ga >> 32) & 0x01ffffffu) | (2u << 30) };
        // D# Group 1: data_size=4B(2), iterate_enable=1; tensor_dim0=2048;
        // tensor_dim1=1; tile_dim0=2048; tile_dim1=1; dim0_stride=2048.
        v8i g1 = { (int)((2u << 16) | (1u << 19)),
                   (int)(2048u << 16),   // tensor_dim0[15:0] -> bits[63:48]
                   (int)(1u << 16),      // tensor_dim0[31:16]=0; tensor_dim1[15:0]=1
                   (int)(2048u << 16),   // tensor_dim1[31:16]=0; tile_dim0=2048
                   (int)1,               // tile_dim1=1; tile_dim2=0
                   (int)2048,            // tensor_dim0_stride[31:0]
                   0,                    // stride0[47:32]=0; tensor_dim1_stride lo16=0
                   0 };
        // D# Group 2 (iterate mode): lds_inc=2056 elems, global_inc=2048 elems,
        // iterate_count=31 (-> 32 rows).
        v4i g2  = { 0, (int)W_PITCH, (int)H_DIM, (int)(31u << 16) };
        v4i g3  = { 0, 0, 0, 0 };
        v8i aux = { 0, 0, 0, 0, 0, 0, 0, 0 };
        __builtin_amdgcn_tensor_load_to_lds(g0, g1, g2, g3, aux, 0);
        __builtin_amdgcn_s_wait_tensorcnt(0);
    }

    const int   r     = tid >> 3;         // 0..31: output row within slice
    const int   sub   = tid & 7;          // 8 threads cooperate per row
    const int   row_g = wg * ROWS_PER_WG + r;
    const float bias  = bhh[row_g];

    for (int t = 0; t < T_DIM; ++t) {
        // Broadcast h_{t-1} into LDS (h0 at t==0, else global double buffer).
        const float* hsrc = (t == 0) ? h0 : (hbuf + (size_t)(t & 1) * H_DIM);
        for (int i = tid; i < H_DIM; i += RNN_THREADS) hs[i] = hsrc[i];
        __syncthreads();   // also orders the one-time W preload at t==0

        // Partial dot: thread covers k = sub, sub+8, ... (distinct banks).
        float acc = 0.f;
        const float* wr = wsl + r * W_PITCH + sub;
        const float* hp = hs + sub;
#pragma unroll 8
        for (int i = 0; i < H_DIM / 8; ++i)
            acc += wr[i * 8] * hp[i * 8];

        red[tid] = acc;
        __syncthreads();

        if (sub == 0) {
            float s = red[tid] + red[tid + 1] + red[tid + 2] + red[tid + 3]
                    + red[tid + 4] + red[tid + 5] + red[tid + 6] + red[tid + 7];
            float v = tanhf(xpo[(size_t)t * H_DIM + row_g] + s + bias);
            xpo[(size_t)t * H_DIM + row_g] = v;                  // outs[t]
            hbuf[(size_t)((t + 1) & 1) * H_DIM + row_g] = v;     // next h
        }
        __syncthreads();
        __threadfence();   // make h_new visible device-wide before arrive

        // Grid barrier: monotonic arrive counter, acquire-spin by thread 0.
        if (tid == 0) {
            __hip_atomic_fetch_add(counter, 1u, __ATOMIC_ACQ_REL,
                                   __HIP_MEMORY_SCOPE_AGENT);
            const unsigned target = (unsigned)NWG * (unsigned)(t + 1);
            while (__hip_atomic_load(counter, __ATOMIC_ACQUIRE,
                                     __HIP_MEMORY_SCOPE_AGENT) < target)
                __builtin_amdgcn_s_sleep(1);
        }
        __syncthreads();
    }
}

__global__ void init_counter(unsigned* c) { *c = 0u; }

// ---------------------------------------------------------------------------
// Inputs (setup_inputs order):
// 0 input_values (T,1,H) f32   1 W_ih (H,H)   2 W_hh (H,H)   3 b_ih (H)
// 4 b_hh (H)                   5 W_lin (O,H)  6 b_lin (O)    7 h0 (1,1,H)
// Output: (T,1,O) f32.
// Workspace: [0,32MB) xp/outs (T*H f32) | 2*H f32 h double-buffer | counter.
// ---------------------------------------------------------------------------
extern "C" void kernel_launch(void* const* d_in, const int* in_sizes, int n_in,
                              void* d_out, int out_size, void* d_ws, size_t ws_size,
                              hipStream_t stream)
{
    const float* X    = (const float*)d_in[0];
    const float* Wih  = (const float*)d_in[1];
    const float* Whh  = (const float*)d_in[2];
    const float* bih  = (const float*)d_in[3];
    const float* bhh  = (const float*)d_in[4];
    const float* Wlin = (const float*)d_in[5];
    const float* blin = (const float*)d_in[6];
    const float* h0   = (const float*)d_in[7];
    float*       out  = (float*)d_out;

    char*     ws      = (char*)d_ws;
    float*    xpo     = (float*)ws;                                   // T*H
    float*    hbuf    = (float*)(ws + (size_t)T_DIM * H_DIM * 4);     // 2*H
    unsigned* counter = (unsigned*)(ws + (size_t)T_DIM * H_DIM * 4 + 2 * H_DIM * 4);

    // Reset grid-sync counter every call (ws is not re-poisoned between replays).
    init_counter<<<1, 1, 0, stream>>>(counter);

    // x_proj = X * W_ih^T + b_ih   (WMMA fp32, 32x32 tiles/wave)
    gemm16x16x4<<<dim3(T_DIM / 32, H_DIM / 128), 128, 0, stream>>>(
        X, Wih, bih, xpo, T_DIM, H_DIM, H_DIM);

    // Sequential recurrence, persistent weights in LDS (TDM preload).
    rnn_persistent<<<NWG, RNN_THREADS, RNN_LDS_BYTES, stream>>>(
        xpo, Whh, bhh, h0, hbuf, counter);

    // out = outs * W_lin^T + b_lin (WMMA fp32, 32x32 tiles/wave)
    gemm16x16x4<<<dim3(T_DIM / 32, O_DIM / 128), 128, 0, stream>>>(
        xpo, Wlin, blin, out, T_DIM, O_DIM, H_DIM);
}